// GlobalMultimaxPool1d_41618233099092
// MI455X (gfx1250) — compile-verified
//
#include <hip/hip_runtime.h>
#include <stdint.h>

// Problem constants (x: (16, 256, 2048) f32; output: (16, 256, 16) f32)
constexpr int kN       = 2048;   // row length
constexpr int kOut     = 16;     // GlobalMultimaxPool1d(outputs=16)
constexpr int kThreads = 256;    // 8 wave32 waves per workgroup
constexpr float kEps   = 0.1f;   // torchsort regularization_strength

// Safe no-pooling threshold: float-computed w_i - w_{i+1} is 10.0 +/- ~4e-3,
// always > 9.99. If (max-min) < 9.99 no adjacent sorted gap can violate.
constexpr float kRangeGate = 9.99f;
constexpr float kNegInf    = -3.402823466e38f;

// w_i = (N - i)/eps, exactly as jnp.arange(n, 0, -1)/eps in float32
__device__ __forceinline__ float w_of(int i) {
    return (float)(kN - i) / kEps;
}

__global__ __launch_bounds__(kThreads)
void multimax_softsort_kernel(const float* __restrict__ x,
                              float* __restrict__ out,
                              int rows) {
    __shared__ __align__(16) float a[kN];   // staged row (kept intact on fast path)
    __shared__ float bsum[kN];              // tournament ping / PAV block sums
    __shared__ float bcnt[kN];              // tournament pong / PAV block counts
    __shared__ float rmin[kThreads / 32];
    __shared__ float rmax[kThreads / 32];
    __shared__ int   need_pav;

    const int row = blockIdx.x;
    if (row >= rows) return;                // block-uniform
    const int t    = threadIdx.x;
    const int lane = t & 31;
    const int wid  = t >> 5;

    // ---------------- Stage row into LDS: CDNA5 async global->LDS DMA ----------
    {
        const uint64_t gbase = (uint64_t)(uintptr_t)(x + (size_t)row * kN);
        uint32_t lds0 = (uint32_t)(uintptr_t)(&a[t * 4]);
        uint32_t lds1 = (uint32_t)(uintptr_t)(&a[kN / 2 + t * 4]);
        uint32_t off0 = (uint32_t)(t * 16);
        uint32_t off1 = (uint32_t)(kN / 2 * 4 + t * 16);
        asm volatile("global_load_async_to_lds_b128 %0, %1, %2"
                     :: "v"(lds0), "v"(off0), "s"(gbase) : "memory");
        asm volatile("global_load_async_to_lds_b128 %0, %1, %2"
                     :: "v"(lds1), "v"(off1), "s"(gbase) : "memory");
        asm volatile("s_wait_asynccnt 0" ::: "memory");
    }
    __syncthreads();

    // ---------------- Grab 8 elements/thread (bank-conflict-free, strided) -----
    float r[8];
    float lmin = a[t], lmax = a[t];
#pragma unroll
    for (int q = 0; q < 8; ++q) {
        const float v = a[t + q * kThreads];
        r[q] = v;
        lmin = fminf(lmin, v);
        lmax = fmaxf(lmax, v);
    }

    // ---------------- Row range: wave32 shuffle reduce + tiny LDS reduce -------
#pragma unroll
    for (int m = 16; m >= 1; m >>= 1) {
        lmin = fminf(lmin, __shfl_xor(lmin, m, 32));
        lmax = fmaxf(lmax, __shfl_xor(lmax, m, 32));
    }
    if (lane == 0) { rmin[wid] = lmin; rmax[wid] = lmax; }
    __syncthreads();
    if (t == 0) {
        float mn = rmin[0], mx = rmax[0];
#pragma unroll
        for (int wv = 1; wv < kThreads / 32; ++wv) {
            mn = fminf(mn, rmin[wv]);
            mx = fmaxf(mx, rmax[wv]);
        }
        need_pav = (mx - mn >= kRangeGate) ? 1 : 0;
    }
    __syncthreads();

    if (need_pav == 0) {
        // ================= FAST PATH: exact top-16 tournament =================
        // Descending 8-sorting network (Batcher odd-even merge, 19 CEs).
        auto ce = [&](int i, int j) {
            const float hi = fmaxf(r[i], r[j]);
            const float lo = fminf(r[i], r[j]);
            r[i] = hi; r[j] = lo;
        };
        ce(0,1); ce(2,3); ce(4,5); ce(6,7);
        ce(0,2); ce(1,3); ce(4,6); ce(5,7);
        ce(1,2); ce(5,6);
        ce(0,4); ce(1,5); ce(2,6); ce(3,7);
        ce(2,4); ce(3,5);
        ce(1,2); ce(3,4); ce(5,6);

        // Each thread owns a descending run of 8.
#pragma unroll
        for (int q = 0; q < 8; ++q) bsum[t * 8 + q] = r[q];
        __syncthreads();

        // Merge-top-16 tournament: runs halve each round; top-16 is lossless
        // (any globally top-16 element is top-16 of every merge containing it).
        auto merge_round = [&](const float* S, float* D, int runs, int inl) {
            if (t < runs) {
                const float* A = S + (2 * t) * inl;
                const float* B = A + inl;
                float* Dst = D + t * kOut;
                int ia = 0, ib = 0;
#pragma unroll
                for (int e = 0; e < kOut; ++e) {
                    const float va = (ia < inl) ? A[ia] : kNegInf;
                    const float vb = (ib < inl) ? B[ib] : kNegInf;
                    if (va >= vb) { Dst[e] = va; ++ia; }
                    else          { Dst[e] = vb; ++ib; }
                }
            }
            __syncthreads();
        };
        merge_round(bsum, bcnt, 128, 8);
        merge_round(bcnt, bsum,  64, 16);
        merge_round(bsum, bcnt,  32, 16);
        merge_round(bcnt, bsum,  16, 16);
        merge_round(bsum, bcnt,   8, 16);
        merge_round(bcnt, bsum,   4, 16);
        merge_round(bsum, bcnt,   2, 16);
        merge_round(bcnt, bsum,   1, 16);   // final descending top-16 in bsum[0..15]

        if (t < kOut) out[(size_t)row * kOut + t] = bsum[t];
        return;
    }

    // ================= RARE FALLBACK: full sort + exact PAV ====================
    // a[] is still the untouched row. Bitonic sort ascending in LDS.
    for (int k = 2; k <= kN; k <<= 1) {
        for (int j = k >> 1; j > 0; j >>= 1) {
#pragma unroll
            for (int q = 0; q < (kN / 2) / kThreads; ++q) {
                const int p   = t + q * kThreads;
                const int i   = ((p & ~(j - 1)) << 1) | (p & (j - 1));
                const int ixj = i | j;
                const bool up = ((i & k) == 0);
                const float vi = a[i];
                const float vj = a[ixj];
                if ((vi > vj) == up) { a[i] = vj; a[ixj] = vi; }
            }
            __syncthreads();
        }
    }

    // Reference-identical PAV on y_i = w_i + a_i (single thread; rare path).
    if (t == 0) {
        int top = -1;
        for (int i = 0; i < kN; ++i) {
            const float y = w_of(i) + a[i];
            ++top;
            bsum[top] = y;
            bcnt[top] = 1.0f;
            while (top > 0 && bsum[top] * bcnt[top - 1] > bsum[top - 1] * bcnt[top]) {
                bsum[top - 1] += bsum[top];
                bcnt[top - 1] += bcnt[top];
                --top;
            }
        }
        int pos = kN;
        for (int b = top; b >= 0 && pos > kN - kOut; --b) {
            const int start = pos - (int)bcnt[b];
            const float mean = bsum[b] / bcnt[b];
            const int lo = (start > kN - kOut) ? start : (kN - kOut);
            for (int p = lo; p < pos; ++p) {
                out[(size_t)row * kOut + (kN - 1 - p)] = mean - w_of(p);
            }
            pos = start;
        }
    }
}

extern "C" void kernel_launch(void* const* d_in, const int* in_sizes, int n_in,
                              void* d_out, int out_size, void* d_ws, size_t ws_size,
                              hipStream_t stream) {
    (void)n_in; (void)out_size; (void)d_ws; (void)ws_size;
    const float* x = (const float*)d_in[0];
    float* out = (float*)d_out;
    const int rows = in_sizes[0] / kN;   // 16*256 = 4096 rows
    multimax_softsort_kernel<<<dim3(rows), dim3(kThreads), 0, stream>>>(x, out, rows);
}